// RECT_L_2714419331272
// MI455X (gfx1250) — compile-verified
//
#include <hip/hip_runtime.h>
#include <hip/hip_bf16.h>

// ---------------------------------------------------------------------------
// GCNConv (256 -> 128) + Linear (128 -> 256) on gfx1250.
// Matmuls: V_WMMA_F32_16X16X4_F32, one 16x64 tile per wave (4 accumulators,
// A fragment reused across 4 WMMAs; all B fragments preloaded per K-step so
// the 4 WMMAs issue back-to-back behind one loadcnt wait).
// Edge phase is L2-resident f32 atomics (h and g both fit in the 192MB L2).
// ---------------------------------------------------------------------------

typedef float v2f __attribute__((ext_vector_type(2)));
typedef float v8f __attribute__((ext_vector_type(8)));

// ---- degree / normalization -----------------------------------------------

__global__ void deg_init_kernel(float* __restrict__ deg, int n) {
    int i = blockIdx.x * blockDim.x + threadIdx.x;
    if (i < n) deg[i] = 1.0f;                 // self-loop contributes 1
}

__global__ void deg_count_kernel(const int* __restrict__ dst,
                                 float* __restrict__ deg, int E) {
    int e = blockIdx.x * blockDim.x + threadIdx.x;
    if (e < E) atomicAdd(&deg[dst[e]], 1.0f);
}

__global__ void dinv_kernel(float* __restrict__ deg, int n) {
    int i = blockIdx.x * blockDim.x + threadIdx.x;
    if (i < n) deg[i] = rsqrtf(deg[i]);       // deg >= 1 always (self-loop)
}

// ---- fp32 WMMA GEMM: C[M,Nc] = A[M,K] @ B[K,Nc] (+ bias per column) -------
// One wave computes a 16x64 tile: 4 accumulators along N sharing one A
// fragment per K-step. Requires M%16==0, Nc%64==0, K%4==0; all guards are
// wave-uniform so EXEC is all-ones at every WMMA.

__global__ void gemm_wmma_kernel(const float* __restrict__ A,
                                 const float* __restrict__ B,
                                 const float* __restrict__ bias,
                                 float* __restrict__ C,
                                 int M, int Nc, int K) {
    const int lane   = threadIdx.x & 31;
    const int wave   = blockIdx.x * (blockDim.x >> 5) + (threadIdx.x >> 5);
    const int tilesN = Nc >> 6;               // 64-wide N tiles
    const int tm     = wave / tilesN;
    const int tn     = wave - tm * tilesN;
    if (tm * 16 >= M) return;                 // wave-uniform exit only

    const int row     = tm * 16 + (lane & 15);  // A row this lane feeds
    const int colBase = tn * 64 + (lane & 15);  // first of 4 B/C columns
    const int kb      = (lane >> 4) << 1;       // lanes 0-15 -> K+0/1, 16-31 -> K+2/3

    // C/D layout: VGPR r holds row M = tm*16 + r + 8*(lane>=16), col = colBase+16j.
    // Bias is per-column -> every accumulator slot of tile j starts at bias[col_j].
    v8f c[4];
#pragma unroll
    for (int j = 0; j < 4; ++j) {
        const float bv = bias ? bias[colBase + 16 * j] : 0.0f;
#pragma unroll
        for (int r = 0; r < 8; ++r) c[j][r] = bv;
    }

    const float* Arow = A + (size_t)row * K + kb;   // 8B-aligned (kb even, K%4==0)
    const float* Bp   = B + (size_t)kb * Nc + colBase;

#pragma unroll 4
    for (int k = 0; k < K; k += 4) {
        // Pull the streaming A operand into all cache levels (WGP scope).
        if ((k & 15) == 0) __builtin_prefetch(Arow + k + 64, 0, 3);

        // Issue all loads for this K-step first (one clause), then 4 WMMAs
        // back-to-back behind a single loadcnt wait.
        v2f a = *(const v2f*)(Arow + k);            // two consecutive K values
        const float* b0 = Bp + (size_t)k * Nc;
        v2f b[4];
#pragma unroll
        for (int j = 0; j < 4; ++j) {
            b[j].x = b0[j * 16];                    // row k+kb,   col colBase+16j
            b[j].y = b0[j * 16 + Nc];               // row k+kb+1, same col
        }
#pragma unroll
        for (int j = 0; j < 4; ++j) {
            // (neg_a, A, neg_b, B, c_mod, C, reuse_a, reuse_b)
            c[j] = __builtin_amdgcn_wmma_f32_16x16x4_f32(false, a, false, b[j],
                                                         (short)0, c[j],
                                                         false, false);
        }
    }

    const int mbase = tm * 16 + ((lane >> 4) << 3);
#pragma unroll
    for (int j = 0; j < 4; ++j) {
        float* Crow = C + (size_t)mbase * Nc + colBase + 16 * j;
#pragma unroll
        for (int r = 0; r < 8; ++r) Crow[(size_t)r * Nc] = c[j][r];
    }
}

// ---- self-loop message + conv bias: g = b_conv + h * dinv^2 ---------------

__global__ void self_bias_kernel(const float* __restrict__ h,
                                 const float* __restrict__ dinv,
                                 const float* __restrict__ bconv,
                                 float* __restrict__ g, int n, int hid) {
    int idx = blockIdx.x * blockDim.x + threadIdx.x;
    if (idx >= n * hid) return;
    int i = idx / hid;
    int f = idx - i * hid;
    float di = dinv[i];
    g[idx] = bconv[f] + h[idx] * di * di;
}

// ---- edge scatter: g[dst] += h[src] * dinv[src]*dinv[dst] -----------------
// One wave per edge; each lane moves a float4 (32 lanes * 4 = 128 = hid).
// h and g both fit in the 192MB L2, so gather + atomics stay on-chip.

__global__ void scatter_kernel(const float* __restrict__ h,
                               const int* __restrict__ src,
                               const int* __restrict__ dst,
                               const float* __restrict__ dinv,
                               float* __restrict__ g, int E, int hid) {
    long long t = (long long)blockIdx.x * blockDim.x + threadIdx.x;
    int e = (int)(t >> 5);
    if (e >= E) return;
    int lane = (int)t & 31;
    int s = src[e];
    int d = dst[e];
    float nrm = dinv[s] * dinv[d];
    float4 v = ((const float4*)(h + (size_t)s * hid))[lane];
    float* gp = g + (size_t)d * hid + lane * 4;
    atomicAdd(gp + 0, v.x * nrm);
    atomicAdd(gp + 1, v.y * nrm);
    atomicAdd(gp + 2, v.z * nrm);
    atomicAdd(gp + 3, v.w * nrm);
}

// ---------------------------------------------------------------------------

extern "C" void kernel_launch(void* const* d_in, const int* in_sizes, int n_in,
                              void* d_out, int out_size, void* d_ws, size_t ws_size,
                              hipStream_t stream) {
    const float* x    = (const float*)d_in[0];
    const int*   eidx = (const int*)  d_in[1];
    const float* Wc   = (const float*)d_in[2];
    const float* bc   = (const float*)d_in[3];
    const float* Wl   = (const float*)d_in[4];
    const float* bl   = (const float*)d_in[5];
    float*       out  = (float*)d_out;

    const int hid  = in_sizes[3];          // 128
    const int in_c = in_sizes[5];          // 256
    const int N    = in_sizes[0] / in_c;   // 50000 (divisible by 16)
    const int E    = in_sizes[1] / 2;      // 800000
    const int* src = eidx;                 // edge_index[0]
    const int* dst = eidx + E;             // edge_index[1]

    // workspace: dinv | h | g   (~51.5 MB total)
    char*  ws  = (char*)d_ws;
    size_t off = 0;
    float* dinv = (float*)(ws + off); off += (((size_t)N * 4) + 255) & ~(size_t)255;
    float* h    = (float*)(ws + off); off += (((size_t)N * hid * 4) + 255) & ~(size_t)255;
    float* g    = (float*)(ws + off);

    // 1) degrees (with self-loop) -> dinv = rsqrt(deg)
    deg_init_kernel <<<(N + 255) / 256, 256, 0, stream>>>(dinv, N);
    deg_count_kernel<<<(E + 255) / 256, 256, 0, stream>>>(dst, dinv, E);
    dinv_kernel     <<<(N + 255) / 256, 256, 0, stream>>>(dinv, N);

    // 2) h = x @ W_conv   [N, hid]
    {
        int tiles  = (N / 16) * (hid / 64);            // 6250 waves (16x64 tiles)
        int blocks = (tiles + 7) / 8;                  // 8 waves / 256-thread block
        gemm_wmma_kernel<<<blocks, 256, 0, stream>>>(x, Wc, nullptr, h, N, hid, in_c);
    }

    // 3) g = b_conv + h * dinv^2  (self-loop message + bias)
    self_bias_kernel<<<(N * hid + 255) / 256, 256, 0, stream>>>(h, dinv, bc, g, N, hid);

    // 4) g[dst] += h[src] * dinv[src]*dinv[dst]  (one wave per edge)
    {
        long long work   = (long long)E * 32;
        int       blocks = (int)((work + 255) / 256);
        scatter_kernel<<<blocks, 256, 0, stream>>>(h, src, dst, dinv, g, E, hid);
    }

    // 5) out = g @ W_lin + b_lin   [N, in_c]
    {
        int tiles  = (N / 16) * (in_c / 64);           // 12500 waves
        int blocks = (tiles + 7) / 8;
        gemm_wmma_kernel<<<blocks, 256, 0, stream>>>(g, Wl, bl, out, N, in_c, hid);
    }
}